// GNN_32091995636000
// MI455X (gfx1250) — compile-verified
//
#include <hip/hip_runtime.h>
#include <stdint.h>

typedef __attribute__((ext_vector_type(2))) float v2f;
typedef __attribute__((ext_vector_type(8))) float v8f;
typedef __attribute__((ext_vector_type(4))) int   v4i;

#define N_NODES   50000
#define N_EDGES   800000
#define N_GRAPHS  1024
#define NODE_IN   128
#define HID       300
#define HP        304      // HID padded to multiple of 16
#define HP4       76       // HP / 4
#define DEPTH     5
#define READOUT   1024

// ---------------------------------------------------------------------------
// Async global->LDS staging (ASYNCcnt-tracked). Falls back to synchronous
// float4 staging if the toolchain doesn't declare the gfx1250 builtins.
// Round-2 probe: param 0 of the async builtin is `int4 addrspace(1)*`.
// ---------------------------------------------------------------------------
#if defined(__has_builtin)
#if __has_builtin(__builtin_amdgcn_global_load_async_to_lds_b128) && \
    __has_builtin(__builtin_amdgcn_s_wait_asynccnt)
#define USE_ASYNC 1
#endif
#endif
#ifndef USE_ASYNC
#define USE_ASYNC 0
#endif

typedef __attribute__((address_space(1))) v4i gas_v4i;   // global int4
typedef __attribute__((address_space(3))) v4i las_v4i;   // LDS int4

__device__ __forceinline__ void stage16(const float* __restrict__ g, float* l) {
#if USE_ASYNC
    __builtin_amdgcn_global_load_async_to_lds_b128(
        (gas_v4i*)g, (las_v4i*)l, 0, 0);
#else
    *reinterpret_cast<float4*>(l) = *reinterpret_cast<const float4*>(g);
#endif
}

__device__ __forceinline__ void wait_stage(bool pending_next) {
#if USE_ASYNC
    if (pending_next) __builtin_amdgcn_s_wait_asynccnt(2);  // retire current tile's 2 A-issues
    else              __builtin_amdgcn_s_wait_asynccnt(0);
#else
    (void)pending_next;
#endif
}

// ---------------------------------------------------------------------------
// Padding kernels: copy [Ks x Ns] (batched) into zero-padded [Kp x Np]
// ---------------------------------------------------------------------------
__global__ void pad2d_kernel(float* __restrict__ dst, const float* __restrict__ src,
                             int Kp, int Np, int Ks, int Ns, int nb) {
    int i = blockIdx.x * blockDim.x + threadIdx.x;
    int total = nb * Kp * Np;
    if (i >= total) return;
    int b = i / (Kp * Np);
    int r = (i / Np) % Kp;
    int c = i % Np;
    float v = 0.0f;
    if (r < Ks && c < Ns) v = src[(size_t)b * Ks * Ns + (size_t)r * Ns + c];
    dst[i] = v;
}

__global__ void pad1d_kernel(float* __restrict__ dst, const float* __restrict__ src,
                             int Np, int Ns, int nb) {
    int i = blockIdx.x * blockDim.x + threadIdx.x;
    int total = nb * Np;
    if (i >= total) return;
    int b = i / Np;
    int c = i % Np;
    float v = 0.0f;
    if (c < Ns) v = src[(size_t)b * Ns + c];
    dst[i] = v;
}

// ---------------------------------------------------------------------------
// Tiled fp32 WMMA GEMM:  C = act(A[MxK] * B[KxNn] + bias)
// 256 threads = 8 waves; block tile 128(M) x 128(N), BK = 16.
// Wave tile 32x64 = 2x4 WMMA accumulators (V_WMMA_F32_16X16X4_F32).
// A: double-buffered async global->LDS (ASYNCcnt), row-major [m][k].
// B: register-pipelined global loads + transposed LDS stores [n][k], so BOTH
//    fragment reads are contiguous ds b64 pairs (no VGPR repack movs).
// Staging uses address CLAMPING: clamped garbage only reaches output
// rows/cols the store guard drops. K % 16 == 0, Nn % 8 == 0 by construction.
// ACT: 0 = none, 1 = relu, 2 = prelu(alpha)
// ---------------------------------------------------------------------------
template <int ACT>
__global__ void __launch_bounds__(256)
gemm_wmma_kernel(const float* __restrict__ A, int lda,
                 const float* __restrict__ B, int ldb,
                 const float* __restrict__ bias,
                 float* __restrict__ C, int ldc,
                 int M, int Nn, int K,
                 const float* __restrict__ alpha_ptr) {
    // Row pad 20 floats: 16B-aligned rows (async b128 targets, b64 frag reads)
    // and conflict-free bank spread for 16-lane strided access (gcd(20,64)=4,
    // 20*i mod 64 distinct for i=0..15).
    __shared__ float As[2][128][20];    // A tile, row-major  [m][k]
    __shared__ float Bs[2][128][20];    // B tile, transposed [n][k]

    const int tid  = threadIdx.x;
    const int lane = tid & 31;
    const int wave = tid >> 5;
    const int wm   = wave >> 1;     // 0..3 -> M offset wm*32
    const int wn   = wave & 1;      // 0..1 -> N offset wn*64
    const int lr   = lane & 15;
    const int khi  = lane >> 4;     // 0/1: selects K pair (frags) and M half (C)

    const int m0 = blockIdx.x * 128;
    const int n0 = blockIdx.y * 128;

    // A staging: 2 async 16B chunks per thread.
    const int arow0 = tid >> 2;               // 0..63
    const int arow1 = arow0 + 64;             // 64..127
    const int ac    = (tid & 3) << 2;         // 0,4,8,12
    const int gr0   = min(m0 + arow0, M - 1); // clamp rows
    const int gr1   = min(m0 + arow1, M - 1);

    // B staging: one k-row (tid&15), 8 columns ((tid>>4)*8) per thread.
    const int bkrow = tid & 15;               // k within tile
    const int bcl   = (tid >> 4) << 3;        // local col base: 0,8,...,120
    const int gn    = min(n0 + bcl, Nn - 8);  // clamp cols (Nn % 8 == 0)

    auto stageA = [&](int bufi, int k0) {     // exactly 2 async issues
        stage16(&A[(size_t)gr0 * lda + k0 + ac], &As[bufi][arow0][ac]);
        stage16(&A[(size_t)gr1 * lda + k0 + ac], &As[bufi][arow1][ac]);
    };
    auto loadB = [&](int k0, float4& r0, float4& r1) {
        const float* p = &B[(size_t)(k0 + bkrow) * ldb + gn];
        r0 = *reinterpret_cast<const float4*>(p);
        r1 = *reinterpret_cast<const float4*>(p + 4);
    };
    auto storeB = [&](int bufi, const float4& r0, const float4& r1) {
        Bs[bufi][bcl + 0][bkrow] = r0.x;
        Bs[bufi][bcl + 1][bkrow] = r0.y;
        Bs[bufi][bcl + 2][bkrow] = r0.z;
        Bs[bufi][bcl + 3][bkrow] = r0.w;
        Bs[bufi][bcl + 4][bkrow] = r1.x;
        Bs[bufi][bcl + 5][bkrow] = r1.y;
        Bs[bufi][bcl + 6][bkrow] = r1.z;
        Bs[bufi][bcl + 7][bkrow] = r1.w;
    };

    v8f acc[2][4] = {};

    // Prologue: tile 0 in flight.
    {
        float4 r0, r1;
        loadB(0, r0, r1);
        stageA(0, 0);
        storeB(0, r0, r1);
    }

    int buf = 0;
    for (int k0 = 0; k0 < K; k0 += 16) {
        const bool more = (k0 + 16) < K;
        float4 nb0, nb1;
        if (more) {
            loadB(k0 + 16, nb0, nb1);         // global loads issued, consumed later
            stageA(buf ^ 1, k0 + 16);         // async prefetch of next A tile
        }
        wait_stage(more);                     // current A tile's LDS writes done
        __syncthreads();                      // publishes A async + B ds stores

        #pragma unroll
        for (int kk = 0; kk < 16; kk += 4) {
            const int kb = kk + 2 * khi;      // ISA 16x4 f32 fragment layout
            v2f af[2];
            #pragma unroll
            for (int tm = 0; tm < 2; ++tm) {
                const int r = wm * 32 + tm * 16 + lr;
                af[tm].x = As[buf][r][kb];
                af[tm].y = As[buf][r][kb + 1];
            }
            v2f bf[4];
            #pragma unroll
            for (int tn = 0; tn < 4; ++tn) {
                const int n = wn * 64 + tn * 16 + lr;
                bf[tn].x = Bs[buf][n][kb];
                bf[tn].y = Bs[buf][n][kb + 1];
            }
            #pragma unroll
            for (int tm = 0; tm < 2; ++tm)
                #pragma unroll
                for (int tn = 0; tn < 4; ++tn)
                    acc[tm][tn] = __builtin_amdgcn_wmma_f32_16x16x4_f32(
                        false, af[tm], false, bf[tn],
                        (short)0, acc[tm][tn], false, false);
        }
        __syncthreads();                      // retire reads of buf before overwrite
        if (more) storeB(buf ^ 1, nb0, nb1);  // transposed stores for next tile
        buf ^= 1;
    }

    float alpha = 0.0f;
    if (ACT == 2) alpha = alpha_ptr[0];

    // C 16x16 f32 layout: VGPR v -> M = v (lanes 0-15) / v+8 (lanes 16-31)
    #pragma unroll
    for (int tm = 0; tm < 2; ++tm) {
        const int rbase = m0 + wm * 32 + tm * 16 + 8 * khi;
        #pragma unroll
        for (int tn = 0; tn < 4; ++tn) {
            const int c = n0 + wn * 64 + tn * 16 + lr;
            if (c < Nn) {
                const float bv = bias[c];
                #pragma unroll
                for (int v = 0; v < 8; ++v) {
                    const int r = rbase + v;
                    if (r < M) {
                        float x = acc[tm][tn][v] + bv;
                        if (ACT == 1) x = x > 0.0f ? x : 0.0f;
                        if (ACT == 2) x = x >= 0.0f ? x : alpha * x;
                        C[(size_t)r * ldc + c] = x;
                    }
                }
            }
        }
    }
}

// ---------------------------------------------------------------------------
__global__ void copy4_kernel(float4* __restrict__ dst, const float4* __restrict__ src, int n4) {
    int i = blockIdx.x * blockDim.x + threadIdx.x;
    if (i < n4) dst[i] = src[i];
}

__global__ void zero_kernel(float* __restrict__ p, int n) {
    int i = blockIdx.x * blockDim.x + threadIdx.x;
    if (i < n) p[i] = 0.0f;
}

// agg[dst] += h[src], one thread per (edge, 4-column chunk)
__global__ void edge_agg_kernel(const int* __restrict__ srcI, const int* __restrict__ dstI,
                                const float* __restrict__ h, float* __restrict__ agg,
                                int total) {
    int i = blockIdx.x * blockDim.x + threadIdx.x;
    if (i >= total) return;
    int e = i / HP4;
    int c = (i - e * HP4) * 4;
    int s = srcI[e];
    int d = dstI[e];
    const float4 v = *reinterpret_cast<const float4*>(&h[(size_t)s * HP + c]);
    float* p = &agg[(size_t)d * HP + c];
    atomicAdd(p + 0, v.x);
    atomicAdd(p + 1, v.y);
    atomicAdd(p + 2, v.z);
    atomicAdd(p + 3, v.w);
}

// pooled[batch[n]] += h[n], one thread per (node, 4-column chunk)
__global__ void pool_kernel(const int* __restrict__ batch,
                            const float* __restrict__ h, float* __restrict__ pooled,
                            int total) {
    int i = blockIdx.x * blockDim.x + threadIdx.x;
    if (i >= total) return;
    int n = i / HP4;
    int c = (i - n * HP4) * 4;
    int g = batch[n];
    const float4 v = *reinterpret_cast<const float4*>(&h[(size_t)n * HP + c]);
    float* p = &pooled[(size_t)g * HP + c];
    atomicAdd(p + 0, v.x);
    atomicAdd(p + 1, v.y);
    atomicAdd(p + 2, v.z);
    atomicAdd(p + 3, v.w);
}

// ---------------------------------------------------------------------------
extern "C" void kernel_launch(void* const* d_in, const int* in_sizes, int n_in,
                              void* d_out, int out_size, void* d_ws, size_t ws_size,
                              hipStream_t stream) {
    const float* x      = (const float*)d_in[0];
    const int*   ei     = (const int*)d_in[1];
    const int*   batch  = (const int*)d_in[2];
    const float* W_proj = (const float*)d_in[3];
    const float* b_proj = (const float*)d_in[4];
    const float* W1     = (const float*)d_in[5];
    const float* b1     = (const float*)d_in[6];
    const float* W2     = (const float*)d_in[7];
    const float* b2     = (const float*)d_in[8];
    const float* W_sp   = (const float*)d_in[9];
    const float* b_sp   = (const float*)d_in[10];
    const float* alpha  = (const float*)d_in[11];
    float* out = (float*)d_out;

    const int* e_src = ei;
    const int* e_dst = ei + N_EDGES;

    // ---- carve workspace (floats, 256B aligned chunks) ----
    float* ws = (float*)d_ws;
    size_t off = 0;
    auto carve = [&](size_t n) {
        float* p = ws + off;
        off += (n + 63) & ~(size_t)63;
        return p;
    };
    float* h      = carve((size_t)N_NODES * HP);
    float* agg    = carve((size_t)N_NODES * HP);
    float* mid    = carve((size_t)N_NODES * HP);
    float* pooled = carve((size_t)N_GRAPHS * HP);
    float* Wpp    = carve((size_t)NODE_IN * HP);        // padded W_proj [128 x 304]
    float* W1p    = carve((size_t)DEPTH * HP * HP);     // padded W1 [5 x 304 x 304]
    float* W2p    = carve((size_t)DEPTH * HP * HP);
    float* Wspp   = carve((size_t)HP * READOUT);        // padded W_sp [304 x 1024]
    float* bpp    = carve(HP);
    float* b1p    = carve((size_t)DEPTH * HP);
    float* b2p    = carve((size_t)DEPTH * HP);

    const dim3 blk(256);
    auto nblk = [](int total) { return dim3((unsigned)((total + 255) / 256)); };

    // ---- 1. pad weights & biases (zero pad rows make K=304 exact) ----
    pad2d_kernel<<<nblk(NODE_IN * HP), blk, 0, stream>>>(Wpp, W_proj, NODE_IN, HP, NODE_IN, HID, 1);
    pad2d_kernel<<<nblk(DEPTH * HP * HP), blk, 0, stream>>>(W1p, W1, HP, HP, HID, HID, DEPTH);
    pad2d_kernel<<<nblk(DEPTH * HP * HP), blk, 0, stream>>>(W2p, W2, HP, HP, HID, HID, DEPTH);
    pad2d_kernel<<<nblk(HP * READOUT), blk, 0, stream>>>(Wspp, W_sp, HP, READOUT, HID, READOUT, 1);
    pad1d_kernel<<<nblk(HP), blk, 0, stream>>>(bpp, b_proj, HP, HID, 1);
    pad1d_kernel<<<nblk(DEPTH * HP), blk, 0, stream>>>(b1p, b1, HP, HID, DEPTH);
    pad1d_kernel<<<nblk(DEPTH * HP), blk, 0, stream>>>(b2p, b2, HP, HID, DEPTH);

    const dim3 gemm_grid((N_NODES + 127) / 128, (HP + 127) / 128);   // 391 x 3

    // ---- 2. projection: h = relu(x @ W_proj + b) ----
    gemm_wmma_kernel<1><<<gemm_grid, blk, 0, stream>>>(
        x, NODE_IN, Wpp, HP, bpp, h, HP, N_NODES, HP, NODE_IN, nullptr);

    // ---- 3. GIN layers ----
    const int edge_total = N_EDGES * HP4;
    const int node_total = N_NODES * HP4;
    for (int i = 0; i < DEPTH; ++i) {
        copy4_kernel<<<nblk(node_total), blk, 0, stream>>>(
            (float4*)agg, (const float4*)h, node_total);
        edge_agg_kernel<<<nblk(edge_total), blk, 0, stream>>>(
            e_src, e_dst, h, agg, edge_total);
        gemm_wmma_kernel<1><<<gemm_grid, blk, 0, stream>>>(
            agg, HP, W1p + (size_t)i * HP * HP, HP, b1p + (size_t)i * HP,
            mid, HP, N_NODES, HP, HP, nullptr);
        if (i < DEPTH - 1)
            gemm_wmma_kernel<1><<<gemm_grid, blk, 0, stream>>>(
                mid, HP, W2p + (size_t)i * HP * HP, HP, b2p + (size_t)i * HP,
                h, HP, N_NODES, HP, HP, nullptr);
        else
            gemm_wmma_kernel<0><<<gemm_grid, blk, 0, stream>>>(
                mid, HP, W2p + (size_t)i * HP * HP, HP, b2p + (size_t)i * HP,
                h, HP, N_NODES, HP, HP, nullptr);
    }

    // ---- 4. global add pool ----
    zero_kernel<<<nblk(N_GRAPHS * HP), blk, 0, stream>>>(pooled, N_GRAPHS * HP);
    pool_kernel<<<nblk(node_total), blk, 0, stream>>>(batch, h, pooled, node_total);

    // ---- 5. readout: out = prelu(pooled @ W_sp + b_sp) ----
    const dim3 ro_grid((N_GRAPHS + 127) / 128, (READOUT + 127) / 128);   // 8 x 8
    gemm_wmma_kernel<2><<<ro_grid, blk, 0, stream>>>(
        pooled, HP, Wspp, READOUT, b_sp, out, READOUT, N_GRAPHS, READOUT, HP, alpha);
}